// RefineVitBlock_24644522344952
// MI455X (gfx1250) — compile-verified
//
#include <hip/hip_runtime.h>
#include <hip/hip_bf16.h>
#include <math.h>

// ---------------------------------------------------------------------------
// Problem constants (B=2, C=128, H=W=512, wsz=8 -> 4096 windows, nWF=1228)
// ---------------------------------------------------------------------------
#define Bb    2
#define Cc    128
#define Hh    512
#define Wwid  512
#define NWIN  4096
#define NWF   1228
#define SCALE 0.08838834764831845f   // 128^-0.5

// LDS geometry (padded strides to kill bank conflicts on ds_load_b64 frags)
#define SA   132            // row stride for 64x128 buffers
#define SS   68             // row stride for 64x64 logits buffer
#define BUF  (64 * SA)      // 8448 floats
#define SH_FLOATS (4 * BUF + 64 * SS)   // 38144 floats = 152576 bytes

// workspace layout (in floats; all offsets even -> 8B aligned for b64 loads)
#define SCORE_OFF 0
#define LIST_OFF  8192
#define CNT_OFF   (8192 + 2560)
#define G_OFF     (CNT_OFF + 32)
#define KT_OFF    (G_OFF + 16384)
#define V_OFF     (KT_OFF + 16384)
#define WTQ_OFF   (V_OFF + 16384)
#define WTL_OFF   (WTQ_OFF + 16384)
#define WTP_OFF   (WTL_OFF + 16384)
#define WTKV_OFF  (WTP_OFF + 16384)

typedef __attribute__((ext_vector_type(2))) float v2f;
typedef __attribute__((ext_vector_type(8))) float v8f;

// ---------------------------------------------------------------------------
// WMMA f32 16x16x4 wrapper: D = A(16x4) * B(4x16) + C(16x16), all f32
// ---------------------------------------------------------------------------
__device__ __forceinline__ v8f wmma4(v2f a, v2f b, v8f c) {
  return __builtin_amdgcn_wmma_f32_16x16x4_f32(false, a, false, b, (short)0, c,
                                               false, false);
}

// A fragment: 16x4 tile at (m0,k0). lane<16: M=lane, K=k0,k0+1 ; lane>=16: K+=2
__device__ __forceinline__ v2f ldA(const float* buf, int sa, int m0, int k0,
                                   int lane) {
  int m  = m0 + (lane & 15);
  int kk = k0 + ((lane >> 4) << 1);
  const float* p = buf + m * sa + kk;
  v2f a; a.x = p[0]; a.y = p[1];
  return a;
}

// B fragment.
//  BMODE 0: row-major src[k*sb+n]         (LDS v buffer)
//  BMODE 1: transposed src[n*sb+k]        (LDS k buffer)
//  BMODE 2: packed pairs src[((k>>1)*sb+n)*2 + (k&1)] -> single b64 per lane
template <int BMODE>
__device__ __forceinline__ v2f ldB(const float* src, int sb, int k0, int n0,
                                   int lane) {
  int n  = n0 + (lane & 15);
  int kk = k0 + ((lane >> 4) << 1);      // always even
  v2f b;
  if (BMODE == 0)      { b.x = src[kk * sb + n]; b.y = src[(kk + 1) * sb + n]; }
  else if (BMODE == 1) { b.x = src[n * sb + kk]; b.y = src[n * sb + kk + 1]; }
  else                 { b = *(const v2f*)(src + ((size_t)(kk >> 1) * sb + n) * 2); }
  return b;
}

// C/D 16x16 tile: VGPR r -> (M=m0+r, N=n0+lane) lanes 0-15 ; (M=m0+8+r) lanes 16-31
__device__ __forceinline__ v8f ldC(const float* buf, int sd, int m0, int n0,
                                   int lane) {
  int n  = n0 + (lane & 15);
  int mb = m0 + ((lane >> 4) << 3);
  v8f c;
#pragma unroll
  for (int r = 0; r < 8; ++r) c[r] = buf[(mb + r) * sd + n];
  return c;
}

__device__ __forceinline__ void stD(float* buf, int sd, int m0, int n0,
                                    int lane, v8f d) {
  int n  = n0 + (lane & 15);
  int mb = m0 + ((lane >> 4) << 3);
#pragma unroll
  for (int r = 0; r < 8; ++r) buf[(mb + r) * sd + n] = d[r];
}

// 64xK(A,LDS) x Kx128(B) -> 64x128 D. CMODE: 0 zero, 1 bias bcast, 2 acc from D.
template <int BMODE, int CMODE>
__device__ __forceinline__ void gemm_64x128(const float* A, int sa,
                                            const float* Bm, int sb, float* D,
                                            int sd, int K, const float* bias,
                                            int lane, int wave) {
  int n0 = wave * 16;
  v8f acc[4];
  if (CMODE == 0) {
    v8f z = {0.f, 0.f, 0.f, 0.f, 0.f, 0.f, 0.f, 0.f};
    acc[0] = z; acc[1] = z; acc[2] = z; acc[3] = z;
  } else if (CMODE == 1) {
    float bv = bias[n0 + (lane & 15)];
    v8f z = {bv, bv, bv, bv, bv, bv, bv, bv};
    acc[0] = z; acc[1] = z; acc[2] = z; acc[3] = z;
  } else {
#pragma unroll
    for (int t = 0; t < 4; ++t) acc[t] = ldC(D, sd, t * 16, n0, lane);
  }
  for (int k0 = 0; k0 < K; k0 += 4) {
    v2f bf = ldB<BMODE>(Bm, sb, k0, n0, lane);
#pragma unroll
    for (int t = 0; t < 4; ++t) {
      v2f af = ldA(A, sa, t * 16, k0, lane);
      acc[t] = wmma4(af, bf, acc[t]);
    }
  }
#pragma unroll
  for (int t = 0; t < 4; ++t) stD(D, sd, t * 16, n0, lane, acc[t]);
}

// 64xK x Kx64 -> 64x64 logits (each wave: 2 of the 16 tiles)
template <int BMODE>
__device__ __forceinline__ void gemm_logits(const float* A, int sa,
                                            const float* Bm, int sb, float* S,
                                            int K, int lane, int wave) {
  int n0 = (wave & 3) * 16;
  int m0 = (wave >> 2) * 32;
  v8f z = {0.f, 0.f, 0.f, 0.f, 0.f, 0.f, 0.f, 0.f};
  v8f acc0 = z, acc1 = z;
  for (int k0 = 0; k0 < K; k0 += 4) {
    v2f bf = ldB<BMODE>(Bm, sb, k0, n0, lane);
    v2f a0 = ldA(A, sa, m0, k0, lane);
    v2f a1 = ldA(A, sa, m0 + 16, k0, lane);
    acc0 = wmma4(a0, bf, acc0);
    acc1 = wmma4(a1, bf, acc1);
  }
  stD(S, SS, m0, n0, lane, acc0);
  stD(S, SS, m0 + 16, n0, lane, acc1);
}

__device__ __forceinline__ void softmax64(float* S, int lane, int wave) {
  for (int r = wave; r < 64; r += 8) {
    float x0 = S[r * SS + lane] * SCALE;
    float x1 = S[r * SS + 32 + lane] * SCALE;
    float m = fmaxf(x0, x1);
#pragma unroll
    for (int off = 16; off; off >>= 1) m = fmaxf(m, __shfl_xor(m, off, 32));
    float e0 = expf(x0 - m), e1 = expf(x1 - m);
    float s = e0 + e1;
#pragma unroll
    for (int off = 16; off; off >>= 1) s += __shfl_xor(s, off, 32);
    float inv = 1.0f / s;
    S[r * SS + lane] = e0 * inv;
    S[r * SS + 32 + lane] = e1 * inv;
  }
}

__device__ __forceinline__ float gelu_f(float x) {
  return 0.5f * x * (1.0f + erff(x * 0.70710678118654752f));
}

// ---------------------------------------------------------------------------
// Small prep kernels
// ---------------------------------------------------------------------------
__global__ void init_kernel(int* cnt) {
  if (threadIdx.x < Bb) cnt[threadIdx.x] = 0;
}

__global__ void score_kernel(const float* __restrict__ unc,
                             float* __restrict__ score) {
  int idx = blockIdx.x * blockDim.x + threadIdx.x;
  if (idx >= Bb * NWIN) return;
  int b = idx >> 12, w = idx & (NWIN - 1);
  int h0 = (w >> 6) * 8, w0 = (w & 63) * 8;
  const float* p = unc + ((size_t)b * Hh + h0) * Wwid + w0;
  float s = 0.f;
#pragma unroll
  for (int r = 0; r < 8; ++r)
#pragma unroll
    for (int c = 0; c < 8; ++c) s += p[r * Wwid + c];
  score[idx] = s * (1.0f / 64.0f);
}

// exact top-k set via rank counting (same tie-break as lax.top_k):
// processing is per-window independent, so list order doesn't affect output.
__global__ void select_kernel(const float* __restrict__ score,
                              int* __restrict__ list, int* __restrict__ cnt) {
  int idx = blockIdx.x * blockDim.x + threadIdx.x;
  if (idx >= Bb * NWIN) return;
  int b = idx >> 12, w = idx & (NWIN - 1);
  float my = score[idx];
  const float* sb = score + b * NWIN;
  int rank = 0;
  for (int j = 0; j < NWIN; ++j) {
    float v = sb[j];
    rank += (v > my) || (v == my && j < w);
  }
  if (rank < NWF) {
    int slot = atomicAdd(&cnt[b], 1);
    list[b * NWF + slot] = w;
  }
}

// ---------------------------------------------------------------------------
// Fused pass-through copy + 8x8 global pooling.
// One block per (b, c, p1) strip of 64 rows (128 KB). Each thread's segment
// p2 = (tid&127)>>4 is loop-invariant -> one accumulator, deterministic
// half-wave shfl reduction + tiny LDS reduction. Saves a full 256MB HBM read.
// ---------------------------------------------------------------------------
__global__ __launch_bounds__(256) void copypool_kernel(
    const float* __restrict__ fm, float* __restrict__ out,
    float* __restrict__ g) {
  __shared__ float red[16];
  int blk = blockIdx.x;                    // Bb*Cc*8 = 2048
  int b = blk / (Cc * 8);
  int c = (blk / 8) % Cc;
  int p1 = blk & 7;
  const size_t base = ((size_t)(b * Cc + c) * Hh + p1 * 64) * Wwid;
  const float4* src = (const float4*)(fm + base);
  float4* dst = (float4*)(out + base);
  int tid = threadIdx.x, lane = tid & 31, wave = tid >> 5;

  float acc = 0.f;
  for (int i = tid; i < 64 * 128; i += 256) {   // 8192 float4 per block
    float4 v = src[i];
    dst[i] = v;
    acc += v.x + v.y + v.z + v.w;
  }
  // half-wave reduce (lanes 0-15 / 16-31 are distinct p2 groups)
#pragma unroll
  for (int off = 8; off; off >>= 1) acc += __shfl_xor(acc, off, 32);
  if ((lane & 15) == 0) red[wave * 2 + (lane >> 4)] = acc;
  __syncthreads();
  if (tid < 8) {
    // red[j]: waves 0-3 -> p2=j (even rows), waves 4-7 -> p2=j-8 (odd rows)
    float s = red[tid] + red[tid + 8];
    g[(size_t)(b * 64 + p1 * 8 + tid) * Cc + c] = s * (1.0f / 4096.0f);
  }
}

// k,v = g @ kv_g_w.T ; store both in packed WMMA B-fragment layout:
//   kT (contraction over channel, N=64 keys):  kTp[((ch>>1)*64 + key)*2 + (ch&1)]
//   v  (contraction over key,   N=128 chans):  vp [((key>>1)*128 + ch)*2 + (key&1)]
__global__ void kv_kernel(const float* __restrict__ g,
                          const float* __restrict__ kvw,
                          float* __restrict__ kT, float* __restrict__ v) {
  int idx = blockIdx.x * blockDim.x + threadIdx.x;
  if (idx >= Bb * 64 * 256) return;
  int b = idx / (64 * 256);
  int rem = idx % (64 * 256);
  int p = rem / 256, row = rem % 256;
  const float* gp = g + (b * 64 + p) * Cc;
  const float* wr = kvw + row * Cc;
  float acc = 0.f;
  for (int c = 0; c < Cc; ++c) acc += gp[c] * wr[c];
  if (row < Cc) kT[(size_t)b * 8192 + ((size_t)(row >> 1) * 64 + p) * 2 + (row & 1)] = acc;
  else {
    int ch = row - Cc;
    v[(size_t)b * 8192 + ((size_t)(p >> 1) * Cc + ch) * 2 + (p & 1)] = acc;
  }
}

// transpose weights into packed pair layout: dst[((k>>1)*N + n)*2 + (k&1)] = W[n][k]
__global__ void tw_kernel(const float* __restrict__ qg,
                          const float* __restrict__ lin,
                          const float* __restrict__ proj,
                          const float* __restrict__ qkv, float* wq, float* wl,
                          float* wp, float* wkv) {
  for (int idx = blockIdx.x * blockDim.x + threadIdx.x; idx < 3 * 16384 + 49152;
       idx += gridDim.x * blockDim.x) {
    if (idx < 3 * 16384) {
      int which = idx >> 14;
      int r = idx & 16383;
      int n = r >> 7, k = r & 127;             // coalesced source reads
      const float* src = which == 0 ? qg : (which == 1 ? lin : proj);
      float* dst = which == 0 ? wq : (which == 1 ? wl : wp);
      dst[((size_t)(k >> 1) * Cc + n) * 2 + (k & 1)] = src[n * Cc + k];
    } else {
      int r = idx - 3 * 16384;
      int n = r >> 7, k = r & 127;             // n in 0..383
      wkv[((size_t)(k >> 1) * 384 + n) * 2 + (k & 1)] = qkv[n * Cc + k];
    }
  }
}

// ---------------------------------------------------------------------------
// Main per-window kernel: one block (8 wave32) per selected window
// ---------------------------------------------------------------------------
__global__ __launch_bounds__(256) void window_kernel(
    const float* __restrict__ fm, const int* __restrict__ list,
    const float* __restrict__ kT, const float* __restrict__ vg,
    const float* __restrict__ wq, const float* __restrict__ wl,
    const float* __restrict__ wkv, const float* __restrict__ wp,
    const float* __restrict__ lin_b, const float* __restrict__ proj_b,
    float* __restrict__ out) {
  extern __shared__ float sh[];
  float* wf = sh;                 // 64 x 128 (stride SA)
  float* b1 = sh + BUF;
  float* b2 = sh + 2 * BUF;
  float* b3 = sh + 3 * BUF;
  float* sL = sh + 4 * BUF;       // 64 x 64 (stride SS)

  const int tid = threadIdx.x, lane = tid & 31, wave = tid >> 5;
  const int b = blockIdx.x / NWF;
  const int w = list[blockIdx.x];
  const int h0 = (w >> 6) * 8, w0 = (w & 63) * 8;
  const float* fmb = fm + (size_t)b * Cc * Hh * Wwid;
  const float* kTb = kT + (size_t)b * 8192;
  const float* vb = vg + (size_t)b * 8192;

  // ---- gather window: wf[token=r*8+c][ch] = fm[b][ch][h0+r][w0+c]
  for (int idx = tid; idx < Cc * 8 * 2; idx += 256) {
    int ch = idx >> 4, rem = idx & 15, r = rem >> 1, half = rem & 1;
    const float4 vv = *(const float4*)(fmb + ((size_t)ch * Hh + h0 + r) * Wwid +
                                       w0 + half * 4);
    float* dst = wf + (r * 8 + half * 4) * SA + ch;
    dst[0] = vv.x; dst[SA] = vv.y; dst[2 * SA] = vv.z; dst[3 * SA] = vv.w;
  }
  __syncthreads();

  // ---- cross attention: q = wf @ q_g_w.T
  gemm_64x128<2, 0>(wf, SA, wq, Cc, b1, SA, Cc, nullptr, lane, wave);
  __syncthreads();
  // logits = q @ k^T (packed kT, N=64)
  gemm_logits<2>(b1, SA, kTb, 64, sL, Cc, lane, wave);
  __syncthreads();
  softmax64(sL, lane, wave);
  __syncthreads();
  // wf += attn @ v (packed v, N=128; C-operand = wf -> fused residual)
  gemm_64x128<2, 2>(sL, SS, vb, Cc, wf, SA, 64, nullptr, lane, wave);
  __syncthreads();

  // ---- MLP1: wf += gelu(wf @ linner0_w.T + b)
  gemm_64x128<2, 1>(wf, SA, wl, Cc, b1, SA, Cc, lin_b, lane, wave);
  __syncthreads();
  for (int idx = tid; idx < 64 * Cc; idx += 256) {
    int i = idx >> 7, j = idx & 127;
    wf[i * SA + j] += gelu_f(b1[i * SA + j]);
  }
  __syncthreads();

  // ---- self attention: q,k,v = wf @ qkv0_w.T (packed cols 0/128/256)
  gemm_64x128<2, 0>(wf, SA, wkv, 384, b1, SA, Cc, nullptr, lane, wave);
  gemm_64x128<2, 0>(wf, SA, wkv + 256, 384, b2, SA, Cc, nullptr, lane, wave);
  gemm_64x128<2, 0>(wf, SA, wkv + 512, 384, b3, SA, Cc, nullptr, lane, wave);
  __syncthreads();
  // logits = q @ k^T (k from LDS, transposed access)
  gemm_logits<1>(b1, SA, b2, SA, sL, Cc, lane, wave);
  __syncthreads();
  softmax64(sL, lane, wave);
  __syncthreads();
  // av = attn @ v -> b1 (v from LDS, row-major)
  gemm_64x128<0, 0>(sL, SS, b3, SA, b1, SA, 64, nullptr, lane, wave);
  __syncthreads();
  // reference's swapaxes(1,2).reshape residual: wf[f>>7][f&127] += av[f&63][f>>6]
  for (int f = tid; f < 64 * Cc; f += 256) {
    wf[(f >> 7) * SA + (f & 127)] += b1[(f & 63) * SA + (f >> 6)];
  }
  __syncthreads();

  // ---- MLP2: wf += gelu(wf @ proj_w.T + b)
  gemm_64x128<2, 1>(wf, SA, wp, Cc, b1, SA, Cc, proj_b, lane, wave);
  __syncthreads();
  for (int idx = tid; idx < 64 * Cc; idx += 256) {
    int i = idx >> 7, j = idx & 127;
    wf[i * SA + j] += gelu_f(b1[i * SA + j]);
  }
  __syncthreads();

  // ---- scatter back
  float* outb = out + (size_t)b * Cc * Hh * Wwid;
  for (int idx = tid; idx < Cc * 8 * 2; idx += 256) {
    int ch = idx >> 4, rem = idx & 15, r = rem >> 1, half = rem & 1;
    const float* s = wf + (r * 8 + half * 4) * SA + ch;
    float4 vv;
    vv.x = s[0]; vv.y = s[SA]; vv.z = s[2 * SA]; vv.w = s[3 * SA];
    *(float4*)(outb + ((size_t)ch * Hh + h0 + r) * Wwid + w0 + half * 4) = vv;
  }
}

// ---------------------------------------------------------------------------
extern "C" void kernel_launch(void* const* d_in, const int* in_sizes, int n_in,
                              void* d_out, int out_size, void* d_ws,
                              size_t ws_size, hipStream_t stream) {
  const float* fm    = (const float*)d_in[0];
  const float* unc   = (const float*)d_in[1];
  const float* qgw   = (const float*)d_in[2];
  const float* kvw   = (const float*)d_in[3];
  const float* linw  = (const float*)d_in[4];
  const float* linb  = (const float*)d_in[5];
  const float* qkvw  = (const float*)d_in[6];
  const float* projw = (const float*)d_in[7];
  const float* projb = (const float*)d_in[8];
  float* out = (float*)d_out;
  float* ws  = (float*)d_ws;

  float* score = ws + SCORE_OFF;
  int*   list  = (int*)(ws + LIST_OFF);
  int*   cnt   = (int*)(ws + CNT_OFF);
  float* g     = ws + G_OFF;
  float* kT    = ws + KT_OFF;
  float* vv    = ws + V_OFF;
  float* wq    = ws + WTQ_OFF;
  float* wl    = ws + WTL_OFF;
  float* wp    = ws + WTP_OFF;
  float* wkv   = ws + WTKV_OFF;

  init_kernel<<<1, 32, 0, stream>>>(cnt);
  score_kernel<<<(Bb * NWIN + 255) / 256, 256, 0, stream>>>(unc, score);
  select_kernel<<<(Bb * NWIN + 255) / 256, 256, 0, stream>>>(score, list, cnt);
  copypool_kernel<<<Bb * Cc * 8, 256, 0, stream>>>(fm, out, g);
  kv_kernel<<<(Bb * 64 * 256 + 255) / 256, 256, 0, stream>>>(g, kvw, kT, vv);
  tw_kernel<<<256, 256, 0, stream>>>(qgw, linw, projw, qkvw, wq, wl, wp, wkv);
  window_kernel<<<Bb * NWF, 256, SH_FLOATS * sizeof(float), stream>>>(
      fm, list, kT, vv, wq, wl, wkv, wp, linb, projb, out);
}